// GConv_27676769255847
// MI455X (gfx1250) — compile-verified
//
#include <hip/hip_runtime.h>
#include <hip/hip_bf16.h>

typedef __attribute__((ext_vector_type(2))) float v2f;
typedef __attribute__((ext_vector_type(8))) float v8f;

#define BDIM 128
#define KDIM 64

// ---------------------------------------------------------------------------
// m_T[n][b] = sum_k W[k][n] * x[b][k]     (D = A x B, A = W^T, B = x^T)
// One block = one 16-wide n-tile. 8 waves, one 16-wide b-tile each.
// K = 64 -> 16 chained V_WMMA_F32_16X16X4_F32 per wave.
// ---------------------------------------------------------------------------
__global__ __launch_bounds__(256) void gemm_wmma_f32(const float* __restrict__ x,
                                                     const float* __restrict__ w,
                                                     float* __restrict__ mT,
                                                     int N) {
    __shared__ float xs[BDIM * (KDIM + 1)];  // x staged, padded: bank-conflict free
    __shared__ float wsh[KDIM * 16];         // W tile: 64 rows x 16 cols

    const int tid = threadIdx.x;
    const int n0  = blockIdx.x * 16;

    // Stage x[128][64] -> LDS (coalesced read of 8192 contiguous floats)
    for (int i = tid; i < BDIM * KDIM; i += 256) {
        int b = i / KDIM, k = i % KDIM;
        xs[b * (KDIM + 1) + k] = x[i];
    }
    // Stage W tile: wsh[k*16 + c] = W[k][n0+c]  (64B coalesced per row chunk)
    for (int i = tid; i < KDIM * 16; i += 256) {
        int k = i / 16, c = i % 16;
        int n = n0 + c;
        wsh[i] = (n < N) ? w[(size_t)k * N + n] : 0.0f;
    }
    __syncthreads();

    const int wave = tid >> 5;        // 0..7 -> b-tile index
    const int lane = tid & 31;
    const int half = lane >> 4;       // 0 or 1
    const int lm   = lane & 15;
    const int b0   = wave * 16;

    v8f acc = {};
    #pragma unroll
    for (int kk = 0; kk < KDIM; kk += 4) {
        const int ka = kk + 2 * half;
        v2f a, bf;
        // A (16x4 f32): lanes 0-15 K={kk,kk+1}, lanes 16-31 K={kk+2,kk+3}
        a.x = wsh[ka * 16 + lm];
        a.y = wsh[(ka + 1) * 16 + lm];
        // B (4x16 f32): column = lm, rows split across halves same as A
        bf.x = xs[(b0 + lm) * (KDIM + 1) + ka];
        bf.y = xs[(b0 + lm) * (KDIM + 1) + ka + 1];
        // (neg_a, A, neg_b, B, c_mod, C, reuse_a, reuse_b)
        acc = __builtin_amdgcn_wmma_f32_16x16x4_f32(false, a, false, bf,
                                                    (short)0, acc, false, false);
    }

    // D (16x16 f32): acc[r] = D[r + 8*half][lm] -> mT row n0+r+8*half, col b0+lm
    #pragma unroll
    for (int r = 0; r < 8; ++r) {
        int n = n0 + r + 8 * half;
        if (n < N) mT[(size_t)n * BDIM + b0 + lm] = acc[r];
    }
}

// ---------------------------------------------------------------------------
// One wave per edge: gather 128 contiguous floats of m_T[row], scale, and
// atomically add into 128 contiguous floats of out_T[col] (2 cachelines).
// ---------------------------------------------------------------------------
__global__ __launch_bounds__(256) void scatter_edges_T(const float* __restrict__ mT,
                                                       const int* __restrict__ rows,
                                                       const int* __restrict__ cols,
                                                       const float* __restrict__ vals,
                                                       float* __restrict__ outT,
                                                       int nnz) {
    const int e = blockIdx.x * 8 + (threadIdx.x >> 5);
    if (e >= nnz) return;
    const int lane = threadIdx.x & 31;
    const int r = rows[e];
    const int c = cols[e];
    const float v = vals[e];

    const float4* src = (const float4*)(mT + (size_t)r * BDIM);
    float4 d = src[lane];                       // coalesced 512B per wave (L2 hit)
    float* dst = outT + (size_t)c * BDIM + lane * 4;
    atomicAdd(dst + 0, d.x * v);
    atomicAdd(dst + 1, d.y * v);
    atomicAdd(dst + 2, d.z * v);
    atomicAdd(dst + 3, d.w * v);
}

// Fallback: scatter directly into row-major out[b][n] (used only if ws too small)
__global__ __launch_bounds__(256) void scatter_edges_direct(const float* __restrict__ mT,
                                                            const int* __restrict__ rows,
                                                            const int* __restrict__ cols,
                                                            const float* __restrict__ vals,
                                                            float* __restrict__ out,
                                                            int nnz, int N) {
    const int e = blockIdx.x * 8 + (threadIdx.x >> 5);
    if (e >= nnz) return;
    const int lane = threadIdx.x & 31;
    const int r = rows[e];
    const int c = cols[e];
    const float v = vals[e];
    const float4* src = (const float4*)(mT + (size_t)r * BDIM);
    float4 d = src[lane];
    const int b = lane * 4;
    atomicAdd(out + (size_t)(b + 0) * N + c, d.x * v);
    atomicAdd(out + (size_t)(b + 1) * N + c, d.y * v);
    atomicAdd(out + (size_t)(b + 2) * N + c, d.z * v);
    atomicAdd(out + (size_t)(b + 3) * N + c, d.w * v);
}

// ---------------------------------------------------------------------------
// out[b][n] = out_T[n][b]  — LDS-tiled 32x32 transpose
// ---------------------------------------------------------------------------
__global__ __launch_bounds__(256) void transpose_out(const float* __restrict__ outT,
                                                     float* __restrict__ out,
                                                     int N) {
    __shared__ float tile[32][33];
    const int n0 = blockIdx.x * 32;
    const int b0 = blockIdx.y * 32;
    const int tx = threadIdx.x & 31;
    const int ty = threadIdx.x >> 5;   // 0..7

    for (int i = ty; i < 32; i += 8) {
        int n = n0 + i;
        tile[i][tx] = (n < N) ? outT[(size_t)n * BDIM + b0 + tx] : 0.0f;
    }
    __syncthreads();
    for (int i = ty; i < 32; i += 8) {
        int n = n0 + tx;
        if (n < N) out[(size_t)(b0 + i) * N + n] = tile[tx][i];
    }
}

extern "C" void kernel_launch(void* const* d_in, const int* in_sizes, int n_in,
                              void* d_out, int out_size, void* d_ws, size_t ws_size,
                              hipStream_t stream) {
    const float* x    = (const float*)d_in[0];   // [128, 64]
    const float* w    = (const float*)d_in[1];   // [64, N]
    const int*   rows = (const int*)d_in[2];     // [NNZ]
    const int*   cols = (const int*)d_in[3];     // [NNZ]
    const float* vals = (const float*)d_in[4];   // [NNZ]
    float* out = (float*)d_out;                  // [128, N]

    const int N   = in_sizes[1] / KDIM;
    const int nnz = in_sizes[2];

    const size_t mT_bytes = (size_t)N * BDIM * sizeof(float);
    float* mT   = (float*)d_ws;
    float* outT = (float*)((char*)d_ws + mT_bytes);
    const bool use_outT = (ws_size >= 2 * mT_bytes);

    // GEMM: m_T = W^T x^T via fp32 WMMA
    const int ntiles = (N + 15) / 16;
    gemm_wmma_f32<<<ntiles, 256, 0, stream>>>(x, w, mT, N);

    const int egrid = (nnz + 7) / 8;
    if (use_outT) {
        hipMemsetAsync(outT, 0, mT_bytes, stream);
        scatter_edges_T<<<egrid, 256, 0, stream>>>(mT, rows, cols, vals, outT, nnz);
        dim3 tg((N + 31) / 32, BDIM / 32);
        transpose_out<<<tg, 256, 0, stream>>>(outT, out, N);
    } else {
        hipMemsetAsync(out, 0, (size_t)out_size * sizeof(float), stream);
        scatter_edges_direct<<<egrid, 256, 0, stream>>>(mT, rows, cols, vals, out, nnz, N);
    }
}